// NeuralNetwork_17944373363300
// MI455X (gfx1250) — compile-verified
//
#include <hip/hip_runtime.h>
#include <hip/hip_bf16.h>

typedef float v2f __attribute__((ext_vector_type(2)));
typedef float v8f __attribute__((ext_vector_type(8)));

constexpr int Bn  = 4096;
constexpr int Nn  = 2048;
constexpr int NP1 = Nn + 1;

// Branch-free tanh. gfx1250 has hardware V_TANH_F32 (TRANS32 pipe, co-executes
// with WMMA); fall back to an exp2/rcp formulation on |x| (e in (0,1], so no
// overflow/NaN path and no EXEC divergence).
__device__ __forceinline__ float tanh_hw(float x) {
#if __has_builtin(__builtin_amdgcn_tanhf)
    return __builtin_amdgcn_tanhf(x);
#else
    const float ax = __builtin_fabsf(x);
    const float e  = __builtin_amdgcn_exp2f(-2.885390081777927f * ax); // exp(-2|x|)
    const float r  = __builtin_amdgcn_rcpf(1.0f + e);
    const float t  = __builtin_fmaf(-2.0f * e, r, 1.0f);               // tanh(|x|)
    return __builtin_copysignf(t, x);
#endif
}

// One wave (32 lanes) integrates 16 batteries. Batteries sit in the WMMA N
// dimension; weights sit in the A matrix (loaded once, constant over steps).
// All matrix math is V_WMMA_F32_16X16X4_F32 (fp32, matches reference dtype).
__global__ __launch_bounds__(256) void node_scan_wmma(
    const float* __restrict__ thr,  // (B, N+1)
    const float* __restrict__ DOD,  // (1, B)
    const float* __restrict__ Vav,  // (1, B)
    const float* __restrict__ C0,   // (1, B)
    const float* __restrict__ R0,   // (1, B)
    const float* __restrict__ W1,   // (10, 5) row-major
    const float* __restrict__ b1,   // (10,)
    const float* __restrict__ W2,   // (5, 10) row-major
    const float* __restrict__ b2,   // (5,)
    float* __restrict__ out)        // (B, N+1, 2)
{
    const int lane = threadIdx.x & 31;
    const int wave = blockIdx.x * (blockDim.x >> 5) + (threadIdx.x >> 5);
    const int m    = lane & 15;          // battery within tile (N column)
    const bool hi  = lane >= 16;         // high half-wave holds K=2,3 slots
    const int bg   = wave * 16 + m;      // global battery index

    // ---- A-layout weights (constant over the whole scan) -------------------
    // A 16x4 f32 layout: lanes 0-15 hold {K0,K1}, lanes 16-31 hold {K2,K3},
    // row M = lane%16.
    v2f aw1c0, aw1c1;                    // W1 padded to 16x8 (K chunks 0,1)
    {
        const int j  = m;                // hidden row
        const int i0 = hi ? 2 : 0;
        const int i1 = hi ? 3 : 1;
        aw1c0[0] = (j < 10) ? W1[j * 5 + i0] : 0.f;
        aw1c0[1] = (j < 10) ? W1[j * 5 + i1] : 0.f;
        aw1c1[0] = (!hi && j < 10) ? W1[j * 5 + 4] : 0.f;  // K=4; K=5,6,7 = 0
        aw1c1[1] = 0.f;
    }
    v2f aw2c0, aw2c1, aw2c2;             // W2 padded to 16x12 (K chunks 0,1,2)
    {
        const int k = m;                 // output row
        #pragma unroll
        for (int c = 0; c < 3; ++c) {
            const int j0 = 4 * c + (hi ? 2 : 0);
            const int j1 = j0 + 1;
            const float a0 = (k < 5 && j0 < 10) ? W2[k * 10 + j0] : 0.f;
            const float a1 = (k < 5 && j1 < 10) ? W2[k * 10 + j1] : 0.f;
            if      (c == 0) { aw2c0[0] = a0; aw2c0[1] = a1; }
            else if (c == 1) { aw2c1[0] = a0; aw2c1[1] = a1; }
            else             { aw2c2[0] = a0; aw2c2[1] = a1; }
        }
    }

    // C/D 16x16 f32 layout: VGPR v holds row M = v + 8*hi, col N = lane%16.
    v8f c1init, c2init;                  // broadcast biases (padded with 0)
    #pragma unroll
    for (int v = 0; v < 8; ++v) {
        const int j = v + (hi ? 8 : 0);
        c1init[v] = (j < 10) ? b1[j] : 0.f;
        c2init[v] = (j < 5)  ? b2[j] : 0.f;
    }

    // ---- state (canonical copies live in lanes 0-15) -----------------------
    float y3 = C0[bg];
    float y4 = R0[bg];
    const float dod = DOD[bg];
    const float vav = Vav[bg];
    float t = thr[(size_t)bg * NP1];

    float* orow = out + (size_t)bg * (2 * NP1);
    if (!hi) {                            // traj[:,0,:] = init[:,3:5]
        *reinterpret_cast<float2*>(orow) = make_float2(y3, y4);
    }

    for (int s = 0; s < Nn; ++s) {
        const float tn = thr[(size_t)bg * NP1 + s + 1];
        __builtin_prefetch(&thr[(size_t)bg * NP1 + s + 33], 0, 1);
        const float h = tn - t;

        // B chunk layout (4x16 f32): lanes 0-15 hold {K0,K1}, lanes 16-31
        // hold {K2,K3}, col N = lane%16.  Y^T: K = state component.
        const float y3o = __shfl_xor(y3, 16, 32);   // bring y3 to hi half
        v2f bc0, bc1;
        bc0[0] = hi ? vav : t;     // K2=V_av        | K0=t
        bc0[1] = hi ? y3o : dod;   // K3=y3          | K1=DOD
        bc1[0] = hi ? 0.f : y4;    // K6=0           | K4=y4
        bc1[1] = 0.f;              // K7=0           | K5=0

        // Layer 1: Hpre(16x16) = W1_pad x Y^T + b1
        v8f d1 = __builtin_amdgcn_wmma_f32_16x16x4_f32(
            false, aw1c1, false, bc1, (short)0, c1init, false, false);
        d1 = __builtin_amdgcn_wmma_f32_16x16x4_f32(
            false, aw1c0, false, bc0, (short)0, d1, false, false);

        #pragma unroll
        for (int i = 0; i < 8; ++i) d1[i] = tanh_hw(d1[i]);

        // Re-layout H (D layout) into B chunks for layer 2: only half-wave
        // swaps are needed (lane <-> lane^16).
        const float s0 = __shfl_xor(d1[0], 16, 32);  // H[8][m] to lo half
        const float s1 = __shfl_xor(d1[1], 16, 32);  // H[9][m] to lo half
        const float s2 = __shfl_xor(d1[2], 16, 32);  // H[2][m] to hi half
        const float s3 = __shfl_xor(d1[3], 16, 32);  // H[3][m] to hi half
        const float s6 = __shfl_xor(d1[6], 16, 32);  // H[6][m] to hi half
        const float s7 = __shfl_xor(d1[7], 16, 32);  // H[7][m] to hi half

        v2f p0, p1, p2;
        p0[0] = hi ? s2 : d1[0];  p0[1] = hi ? s3 : d1[1];   // K=0..3
        p1[0] = hi ? s6 : d1[4];  p1[1] = hi ? s7 : d1[5];   // K=4..7
        p2[0] = hi ? 0.f : s0;    p2[1] = hi ? 0.f : s1;     // K=8,9 (10,11=0)

        // Layer 2: OUT(16x16) = W2_pad x H + b2
        v8f d2 = __builtin_amdgcn_wmma_f32_16x16x4_f32(
            false, aw2c2, false, p2, (short)0, c2init, false, false);
        d2 = __builtin_amdgcn_wmma_f32_16x16x4_f32(
            false, aw2c1, false, p1, (short)0, d2, false, false);
        d2 = __builtin_amdgcn_wmma_f32_16x16x4_f32(
            false, aw2c0, false, p0, (short)0, d2, false, false);

        // rows M=3,4 of D2 live in VGPRs 3,4 of the low half-wave
        y3 = fmaf(h, d2[3], y3);
        y4 = fmaf(h, d2[4], y4);
        t  = tn;

        if (!hi) {                        // single 8B store (global_store_b64)
            *reinterpret_cast<float2*>(orow + 2 * (s + 1)) = make_float2(y3, y4);
        }
    }
}

extern "C" void kernel_launch(void* const* d_in, const int* in_sizes, int n_in,
                              void* d_out, int out_size, void* d_ws, size_t ws_size,
                              hipStream_t stream) {
    (void)in_sizes; (void)n_in; (void)d_ws; (void)ws_size; (void)out_size;
    const float* thr = (const float*)d_in[0];
    const float* DOD = (const float*)d_in[1];
    const float* Vav = (const float*)d_in[2];
    const float* C0  = (const float*)d_in[3];
    const float* R0  = (const float*)d_in[4];
    const float* W1  = (const float*)d_in[5];
    const float* b1  = (const float*)d_in[6];
    const float* W2  = (const float*)d_in[7];
    const float* b2  = (const float*)d_in[8];
    float* out = (float*)d_out;

    // 4096 batteries / 16 per wave = 256 waves; 8 waves (256 threads) / block.
    dim3 block(256);
    dim3 grid((Bn / 16) / 8);
    node_scan_wmma<<<grid, block, 0, stream>>>(thr, DOD, Vav, C0, R0,
                                               W1, b1, W2, b2, out);
}